// QuantumShadowNoise_84224308674870
// MI455X (gfx1250) — compile-verified
//
#include <hip/hip_runtime.h>
#include <hip/hip_bf16.h>

// ---------------------------------------------------------------------------
// QuantumShadowNoise on MI455X (gfx1250)
//
//   * init layer (RY(asin z0), RZ(asin z1) on all qubits) -> closed form
//   * CNOT-RZ-CNOT entangler blocks are diagonal & commute -> one per-layer
//     65536-entry phase table, fused into the layer's second matmul pass
//   * per-layer RY gates fused 4 qubits at a time into 16x16 REAL matrices,
//     applied pairwise as T' = U_L * T * U_R^T over 16x16 complex tiles via
//     V_WMMA_F32_16X16X4_F32 (4 K-chunks per 16x16x16 product).
//
// Layer pass: one wave per 8 tiles; U fragments register-resident; tile
// buffers are wave-private LDS so ordering uses s_wait_dscnt (no block
// barriers).  State (64MB) is L2-resident (192MB).  6 in-place passes.
// ---------------------------------------------------------------------------

typedef __attribute__((ext_vector_type(2))) float v2f;
typedef __attribute__((ext_vector_type(8))) float v8f;

#define NQ      16
#define NSTATE  65536
#define BATCH   128
#define NLAYERS 3
#define NBASIS  3
#define ZDIM    4
#define WCOLS   31         // 2*NQ - 1
#define SLICES  8          // tiles per wave per layer pass

// Wave-local LDS ordering: DS ops are in-order per wave (ISA 15.5); the asm
// memory clobber stops compiler reordering, s_wait_dscnt 0 drains the queue.
__device__ __forceinline__ void lds_fence_wave() {
  asm volatile("s_wait_dscnt 0x0" ::: "memory");
}

// ---------------------------------------------------------------------------
// WMMA fragment helpers (ISA 7.12.2 layouts, wave32).
// A (16x4 f32, 2 VGPRs): lanes 0-15 / 16-31 both hold M=0..15;
//   VGPR0 holds K=0 (low half) / K=2 (high half); VGPR1 holds K=1 / K=3.
// B (4x16 f32, 2 VGPRs): row k striped across lanes within a VGPR.
// C/D (16x16 f32, 8 VGPRs): VGPR v holds M=v (lanes 0-15) / M=v+8 (16-31),
//   N = lane & 15.
// ---------------------------------------------------------------------------
__device__ __forceinline__ v2f frag_a(const float* M, int k0, int lane) {
  int m = lane & 15, hi = lane >> 4;
  v2f a;
  a.x = M[m * 16 + k0 + (hi ? 2 : 0)];
  a.y = M[m * 16 + k0 + (hi ? 3 : 1)];
  return a;
}
__device__ __forceinline__ v2f frag_b(const float* M, int k0, int lane) {
  int n = lane & 15, hi = lane >> 4;     // B[k][n] = M[k][n]
  v2f b;
  b.x = M[(k0 + 2 * hi + 0) * 16 + n];
  b.y = M[(k0 + 2 * hi + 1) * 16 + n];
  return b;
}
__device__ __forceinline__ v2f frag_bT(const float* M, int k0, int lane) {
  int n = lane & 15, hi = lane >> 4;     // B[k][n] = M[n][k]
  v2f b;
  b.x = M[n * 16 + k0 + 2 * hi + 0];
  b.y = M[n * 16 + k0 + 2 * hi + 1];
  return b;
}

__device__ __forceinline__ v8f wmma4(v2f a, v2f b, v8f c) {
  return __builtin_amdgcn_wmma_f32_16x16x4_f32(false, a, false, b, (short)0, c,
                                               false, false);
}

// P = A * T : A fragments register-resident, T (16x16 row-major) in LDS.
__device__ __forceinline__ v8f mm16_regA(const v2f aF[4], const float* T,
                                         int lane) {
  v8f acc = {0.f, 0.f, 0.f, 0.f, 0.f, 0.f, 0.f, 0.f};
#pragma unroll
  for (int kk = 0; kk < 4; ++kk) acc = wmma4(aF[kk], frag_b(T, 4 * kk, lane), acc);
  return acc;
}
// Q = P * U^T : P (16x16 row-major) in LDS, B=U^T fragments register-resident.
__device__ __forceinline__ v8f mm16_regB(const float* P, const v2f bF[4],
                                         int lane) {
  v8f acc = {0.f, 0.f, 0.f, 0.f, 0.f, 0.f, 0.f, 0.f};
#pragma unroll
  for (int kk = 0; kk < 4; ++kk) acc = wmma4(frag_a(P, 4 * kk, lane), bF[kk], acc);
  return acc;
}

__device__ __forceinline__ void dstore(float* dst, v8f d, int lane) {
  int c = lane & 15, rb = (lane >> 4) * 8;
#pragma unroll
  for (int v = 0; v < 8; ++v) dst[(rb + v) * 16 + c] = d[v];
}

// ---------------------------------------------------------------------------
// Fused 16x16 RY group matrices.  Umat[l][g][r][c], 3*4*256 floats.
// qubit q <-> state bit 15-q; nibble bit 3-j <-> qubit 4g+j. ry=[[c,-s],[s,c]].
// ---------------------------------------------------------------------------
__global__ void qsn_build_u(const float* __restrict__ W, float* __restrict__ Umat) {
  int e = blockIdx.x * 256 + threadIdx.x;     // 0..3071
  int l = e >> 10, g = (e >> 8) & 3, rc = e & 255;
  int r = rc >> 4, c = rc & 15;
  float v = 1.f;
#pragma unroll
  for (int j = 0; j < 4; ++j) {
    float th = W[l * WCOLS + 4 * g + j];
    float ch = __cosf(0.5f * th), sh = __sinf(0.5f * th);
    int rb = (r >> (3 - j)) & 1, cb = (c >> (3 - j)) & 1;
    v *= rb ? (cb ? ch : sh) : (cb ? -sh : ch);
  }
  Umat[e] = v;   // e == ((l*4+g)*16 + r)*16 + c
}

// ---------------------------------------------------------------------------
// Per-layer entangler diagonal: phase[l][i] = exp(i*sum_q s_q th/2),
// s_q = +1 iff bit(15-q) != bit(14-q), th = W[l, 16+q], q = 0..14.
// ---------------------------------------------------------------------------
__global__ void qsn_build_phase(const float* __restrict__ W, float2* __restrict__ phase) {
  int id = blockIdx.x * 256 + threadIdx.x;    // 0 .. 3*65536-1
  int l = id >> 16, i = id & 65535;
  float s = 0.f;
#pragma unroll
  for (int q = 0; q < 15; ++q) {
    float th = 0.5f * W[l * WCOLS + 16 + q];
    int d = ((i >> (15 - q)) ^ (i >> (14 - q))) & 1;
    s += d ? th : -th;
  }
  phase[id] = make_float2(__cosf(s), __sinf(s));
}

// ---------------------------------------------------------------------------
// Closed-form initial state:
// psi[i] = prod_bits (bit? sin : cos)(asin(z0)/2) * exp(i*asin(z1)*(pc-8))
// ---------------------------------------------------------------------------
__global__ void qsn_init(const float* __restrict__ z,
                         float* __restrict__ psiR, float* __restrict__ psiI) {
  unsigned id = blockIdx.x * 256 + threadIdx.x;   // 0 .. 128*65536-1
  int b = id >> 16, i = id & 65535;
  float a0 = asinf(z[2 * b]);
  float t1 = asinf(z[2 * b + 1]);
  float c = __cosf(0.5f * a0), s = __sinf(0.5f * a0);
  float mag = 1.f;
#pragma unroll
  for (int q = 0; q < 16; ++q) mag *= ((i >> q) & 1) ? s : c;
  float ang = t1 * (float)(__popc(i) - 8);
  psiR[id] = mag * __cosf(ang);
  psiI[id] = mag * __sinf(ang);
}

// ---------------------------------------------------------------------------
// Layer pass: one wave handles SLICES consecutive 16x16 complex tiles.
// passB==0: groups (0,1), tile (a,b) over bits 15-12 x 11-8, strided.
// passB==1: groups (2,3), contiguous 256-float tiles; diagonal fused in write.
// Tiles partition the state per pass -> in-place safe.  No block barriers:
// all LDS buffers are wave-private, ordering via s_wait_dscnt.
// ---------------------------------------------------------------------------
__global__ __launch_bounds__(128) void qsn_layer_pass(
    const float* __restrict__ Umat, float* __restrict__ psiR,
    float* __restrict__ psiI, const float2* __restrict__ phase,
    int layer, int passB) {
  __shared__ __align__(16) float T_r[4][256], T_i[4][256], Ps[4][256];

  int tid = threadIdx.x, lane = tid & 31, wave = tid >> 5;
  int gL = passB ? 2 : 0, gR = passB ? 3 : 1;
  const float* UL = Umat + (layer * 4 + gL) * 256;
  const float* UR = Umat + (layer * 4 + gR) * 256;

  // Register-resident operand fragments (invariant across tiles).
  v2f aUL[4], bURT[4];
#pragma unroll
  for (int kk = 0; kk < 4; ++kk) {
    aUL[kk]  = frag_a(UL, 4 * kk, lane);
    bURT[kk] = frag_bT(UR, 4 * kk, lane);
  }

  int wg = blockIdx.x * 4 + wave;           // 0 .. BATCH*256/SLICES - 1
  int b  = wg / (256 / SLICES);
  int t0 = (wg % (256 / SLICES)) * SLICES;

  float* tr = T_r[wave];
  float* ti = T_i[wave];
  float* ps = Ps[wave];
  const float2* ph = phase + (size_t)layer * NSTATE;

#pragma unroll 4
  for (int s = 0; s < SLICES; ++s) {
    int t = t0 + s;
    size_t base;
    int rstride, cstride;
    if (passB) { base = (size_t)b * NSTATE + (size_t)t * 256; rstride = 16;   cstride = 1;   }
    else       { base = (size_t)b * NSTATE + t;               rstride = 4096; cstride = 256; }

    lds_fence_wave();                      // WAR vs previous slice's reads
    if (passB) {
      // contiguous 256-float tile: two b128 loads per lane
      const float4* gr = (const float4*)(psiR + base);
      const float4* gi = (const float4*)(psiI + base);
      float4* lr = (float4*)tr;
      float4* li = (float4*)ti;
      lr[lane]      = gr[lane];
      lr[lane + 32] = gr[lane + 32];
      li[lane]      = gi[lane];
      li[lane + 32] = gi[lane + 32];
      if (s + 1 < SLICES) {                // prefetch next tile
        __builtin_prefetch(psiR + base + 256, 0, 1);
        __builtin_prefetch(psiI + base + 256, 0, 1);
      }
    } else {
#pragma unroll
      for (int j = 0; j < 8; ++j) {
        int e = lane + 32 * j, r = e >> 4, c = e & 15;
        size_t off = base + (size_t)r * rstride + (size_t)c * cstride;
        tr[e] = psiR[off];
        ti[e] = psiI[off];
      }
    }
    lds_fence_wave();                      // tile visible to whole wave

    // Real: Qr = UL * Tr * UR^T   (UL, UR real)
    v8f p = mm16_regA(aUL, tr, lane);
    dstore(ps, p, lane);
    lds_fence_wave();
    v8f qr = mm16_regB(ps, bURT, lane);
    lds_fence_wave();                      // reads of ps done before overwrite

    // Imag: Qi = UL * Ti * UR^T
    v8f pim = mm16_regA(aUL, ti, lane);
    dstore(ps, pim, lane);
    lds_fence_wave();
    v8f qi = mm16_regB(ps, bURT, lane);

#pragma unroll
    for (int v = 0; v < 8; ++v) {
      int r = (lane >> 4) * 8 + v, c = lane & 15;
      size_t off = base + (size_t)r * rstride + (size_t)c * cstride;
      float xr = qr[v], xi = qi[v];
      if (passB) {                         // fused entangler diagonal
        int idx = t * 256 + r * 16 + c;    // full state index
        float2 pc = ph[idx];
        float nr = xr * pc.x - xi * pc.y;
        float ni = xr * pc.y + xi * pc.x;
        xr = nr; xi = ni;
      }
      psiR[off] = xr;
      psiI[off] = xi;
    }
  }
}

// ---------------------------------------------------------------------------
// Pauli-string expectation.  One 256-thread block per (batch, basis).
// P|j> = (-i)^nY * (-1)^{popc(j & sgn)} |j ^ flip>;  <psi|P|psi> is real.
// ---------------------------------------------------------------------------
__global__ __launch_bounds__(256) void qsn_expect(
    const float* __restrict__ psiR, const float* __restrict__ psiI,
    const int* __restrict__ basis, float* __restrict__ qbuf) {
  __shared__ float red[256];
  int job = blockIdx.x;            // 0..383
  int b = job / NBASIS, k = job % NBASIS;

  int flip = 0, sgn = 0, nY = 0;
#pragma unroll
  for (int q = 0; q < 16; ++q) {
    int code = basis[k * 16 + q];  // 0=I 1=X 2=Y 3=Z
    int bit = 1 << (15 - q);
    if (code == 1 || code == 2) flip |= bit;
    if (code == 2 || code == 3) sgn |= bit;
    if (code == 2) nY++;
  }
  float wr, wi;                    // (-i)^nY
  switch (nY & 3) {
    case 0: wr = 1.f;  wi = 0.f;  break;
    case 1: wr = 0.f;  wi = -1.f; break;
    case 2: wr = -1.f; wi = 0.f;  break;
    default: wr = 0.f; wi = 1.f;  break;
  }

  const float* pr = psiR + (size_t)b * NSTATE;
  const float* pi = psiI + (size_t)b * NSTATE;
  float acc = 0.f;
  for (int i = threadIdx.x; i < NSTATE; i += 256) {
    int j = i ^ flip;
    float ar = pr[i], ai = pi[i], br = pr[j], bi = pi[j];
    float rre = ar * br + ai * bi;     // Re[conj(psi_i) * psi_j]
    float rim = ar * bi - ai * br;     // Im[...]
    float sg = (__popc(i & sgn) & 1) ? -1.f : 1.f;
    acc += sg * (wr * rre - wi * rim);
  }
  red[threadIdx.x] = acc;
  __syncthreads();
  for (int st = 128; st > 0; st >>= 1) {
    if (threadIdx.x < st) red[threadIdx.x] += red[threadIdx.x + st];
    __syncthreads();
  }
  if (threadIdx.x == 0) qbuf[job] = red[0];
}

// ---------------------------------------------------------------------------
// out[b,d] = sum_k q[b,k] * proj_w[d,k] + proj_b[d]
// ---------------------------------------------------------------------------
__global__ void qsn_project(const float* __restrict__ qbuf,
                            const float* __restrict__ pw,
                            const float* __restrict__ pb,
                            float* __restrict__ out) {
  int id = blockIdx.x * 128 + threadIdx.x;     // 0..511
  if (id < BATCH * ZDIM) {
    int b = id >> 2, d = id & 3;
    float acc = pb[d];
#pragma unroll
    for (int k = 0; k < NBASIS; ++k) acc += qbuf[b * NBASIS + k] * pw[d * NBASIS + k];
    out[id] = acc;
  }
}

// ---------------------------------------------------------------------------
extern "C" void kernel_launch(void* const* d_in, const int* in_sizes, int n_in,
                              void* d_out, int out_size, void* d_ws, size_t ws_size,
                              hipStream_t stream) {
  const float* z  = (const float*)d_in[0];     // [128,2]
  const float* W  = (const float*)d_in[1];     // [3,31]
  const int*   bs = (const int*)d_in[2];       // [3,16]
  const float* pw = (const float*)d_in[3];     // [4,3]
  const float* pb = (const float*)d_in[4];     // [4]
  float* out = (float*)d_out;                  // [128,4]

  // Workspace layout
  float*  psiR  = (float*)d_ws;                         // 128*65536 f32 (32MB)
  float*  psiI  = psiR + (size_t)BATCH * NSTATE;        // 32MB
  float*  Umat  = psiI + (size_t)BATCH * NSTATE;        // 3*4*256 f32
  float2* phase = (float2*)(Umat + NLAYERS * 4 * 256);  // 3*65536 float2
  float*  qbuf  = (float*)(phase + (size_t)NLAYERS * NSTATE); // 384 f32

  qsn_build_u<<<12, 256, 0, stream>>>(W, Umat);
  qsn_build_phase<<<(NLAYERS * NSTATE) / 256, 256, 0, stream>>>(W, phase);
  qsn_init<<<(BATCH * NSTATE) / 256, 256, 0, stream>>>(z, psiR, psiI);

  const int pass_blocks = (BATCH * 256) / (4 * SLICES);  // 1024
  for (int l = 0; l < NLAYERS; ++l) {
    qsn_layer_pass<<<pass_blocks, 128, 0, stream>>>(Umat, psiR, psiI, phase, l, 0);
    qsn_layer_pass<<<pass_blocks, 128, 0, stream>>>(Umat, psiR, psiI, phase, l, 1);
  }

  qsn_expect<<<BATCH * NBASIS, 256, 0, stream>>>(psiR, psiI, bs, qbuf);
  qsn_project<<<4, 128, 0, stream>>>(qbuf, pw, pb, out);
}